// MDN_MultivariateNormal_18330920419825
// MI455X (gfx1250) — compile-verified
//
#include <hip/hip_runtime.h>
#include <stdint.h>

// Problem constants (from reference)
#define B_ 1024
#define T_ 30
#define M_ 20
#define N_ELEM (B_ * T_ * M_)                 // 614400 (b,t,m) triples
#define CHOL_FLOATS (B_ * M_ * 2*T_ * 2*T_)   // 73,728,000 floats (block-diag target)

// d_out flat offsets (floats), tuple return order:
// probs_traj[B,M], loc[B,M,2T], chol_full[B,M,2T,2T], step_probs[B,T,M],
// mu[B,T,M,2], chol_step[B,T,M,2,2]
#define O_TRAJ     0
#define O_LOC      20480
#define O_CHOL     1249280
#define O_STEP     74977280
#define O_MU       75591680
#define O_CHOLSTEP 76820480

// ---------------------------------------------------------------------------
// CDNA5 async LDS->global store (ASYNCcnt path).
// Builtin signature (probe-confirmed): (v4i AS1*, v4i AS3*, imm offset, imm cpol)
// ---------------------------------------------------------------------------
typedef int v4i __attribute__((vector_size(16)));
typedef __attribute__((address_space(1))) v4i* gv4i_p;
typedef __attribute__((address_space(3))) v4i* lv4i_p;

#if __has_builtin(__builtin_amdgcn_global_store_async_from_lds_b128)
#define ASYNC_ST(gb, lb, imm_off) \
    __builtin_amdgcn_global_store_async_from_lds_b128((gb), (lb), (imm_off), 0)
#else
#define ASYNC_ST(gb, lb, imm_off)                                              \
    asm volatile("global_store_async_from_lds_b128 %0, %1, off offset:%c2"     \
                 :: "v"((uint64_t)(uintptr_t)(gb)),                            \
                    "v"((uint32_t)(uintptr_t)(lb)), "i"(imm_off) : "memory")
#endif

__device__ __forceinline__ void wait_asynccnt0() {
#if __has_builtin(__builtin_amdgcn_s_wait_asynccnt)
    __builtin_amdgcn_s_wait_asynccnt(0);
#else
    asm volatile("s_wait_asynccnt 0" ::: "memory");
#endif
}

// ---------------------------------------------------------------------------
// Kernel 1: stream 295 MB of zeros into chol_full via async LDS->global b128.
// Each block owns a contiguous 80 KB tile: thread i holds ONE base address
// (tile + i*16B) and issues 20 stores at literal immediate offsets k*4096
// (256 threads * 16B per step). Per-issue: 32 lanes x 16B contiguous.
// ---------------------------------------------------------------------------
__global__ void fill_chol_zero(float4* __restrict__ chol) {
    __shared__ float4 zb[64];
    if (threadIdx.x < 64) zb[threadIdx.x] = make_float4(0.f, 0.f, 0.f, 0.f);
    __syncthreads();

    // One base address per thread; block tile = 256*20 float4s = 5120
    size_t base_idx = (size_t)blockIdx.x * 5120 + threadIdx.x;
    gv4i_p gb = (gv4i_p)(v4i*)(chol + base_idx);
    lv4i_p lb = (lv4i_p)(v4i*)(&zb[threadIdx.x & 63]);

    ASYNC_ST(gb, lb, 0 * 4096);
    ASYNC_ST(gb, lb, 1 * 4096);
    ASYNC_ST(gb, lb, 2 * 4096);
    ASYNC_ST(gb, lb, 3 * 4096);
    ASYNC_ST(gb, lb, 4 * 4096);
    ASYNC_ST(gb, lb, 5 * 4096);
    ASYNC_ST(gb, lb, 6 * 4096);
    ASYNC_ST(gb, lb, 7 * 4096);
    ASYNC_ST(gb, lb, 8 * 4096);
    ASYNC_ST(gb, lb, 9 * 4096);
    ASYNC_ST(gb, lb, 10 * 4096);
    ASYNC_ST(gb, lb, 11 * 4096);
    ASYNC_ST(gb, lb, 12 * 4096);
    ASYNC_ST(gb, lb, 13 * 4096);
    ASYNC_ST(gb, lb, 14 * 4096);
    ASYNC_ST(gb, lb, 15 * 4096);
    ASYNC_ST(gb, lb, 16 * 4096);
    ASYNC_ST(gb, lb, 17 * 4096);
    ASYNC_ST(gb, lb, 18 * 4096);
    ASYNC_ST(gb, lb, 19 * 4096);

    wait_asynccnt0();
}

// ---------------------------------------------------------------------------
// Kernel 2: fused per-(b,t,m) transform.
// Reads 5 params, writes mu (float2), chol_step (float4), loc (float2 scatter),
// and the 3 nonzero chol_full entries (on top of the zero fill).
// ---------------------------------------------------------------------------
__global__ void transform_kernel(const float* __restrict__ lat,
                                 float* __restrict__ loc,
                                 float* __restrict__ chol_full,
                                 float* __restrict__ mu,
                                 float* __restrict__ chol_step) {
    int tid = blockIdx.x * blockDim.x + threadIdx.x;  // exact grid: 2400*256
    int m  = tid % M_;
    int bt = tid / M_;
    int t  = bt % T_;
    int b  = bt / T_;

    const float* p = lat + (size_t)tid * 5;
    __builtin_prefetch(p + 5 * 256 * 8, 0, 0);  // global_prefetch_b8, ~8 blocks ahead

    float mux = p[0];
    float muy = p[1];
    float vx  = expf(p[2]) + 1e-6f;
    float vy  = expf(p[3]) + 1e-6f;
    float rho = tanhf(p[4]);
    float sx  = sqrtf(vx);
    float sy  = sqrtf(vy);
    float L11 = sx;
    float L21 = rho * sy;
    float L22 = sy * sqrtf(fmaxf(1.f - rho * rho, 0.f));

    // mu [B,T,M,2] : coalesced float2
    ((float2*)mu)[tid] = make_float2(mux, muy);

    // chol_step [B,T,M,2,2] : rows [L11,0],[L21,L22] -> coalesced float4
    ((float4*)chol_step)[tid] = make_float4(L11, 0.f, L21, L22);

    // loc [B,M,2T] : per-thread float2 at even index (8B aligned)
    size_t locIdx = ((size_t)(b * M_ + m)) * (2 * T_) + 2 * t;
    *(float2*)(loc + locIdx) = make_float2(mux, muy);

    // chol_full nonzeros: row 2t col 2t = L11; row 2t+1 cols (2t,2t+1) = L21,L22
    size_t cb = ((size_t)(b * M_ + m)) * (2 * T_ * 2 * T_) + (size_t)122 * t;
    chol_full[cb] = L11;
    *(float2*)(chol_full + cb + 60) = make_float2(L21, L22);
}

// ---------------------------------------------------------------------------
// Kernel 3/4: softmax -> sigmoid prune -> renorm -> softmax(log(.)) over M=20.
// softmax(log(q)) == q / sum(q); one thread per row, row fits in registers.
// ---------------------------------------------------------------------------
__global__ void prune_softmax_kernel(const float* __restrict__ in,
                                     float* __restrict__ out, int nrows) {
    int r = blockIdx.x * blockDim.x + threadIdx.x;
    if (r >= nrows) return;
    const float* x = in + (size_t)r * M_;
    float v[M_];
    float mx = -INFINITY;
#pragma unroll
    for (int i = 0; i < M_; ++i) { v[i] = x[i]; mx = fmaxf(mx, v[i]); }
    float s = 0.f;
#pragma unroll
    for (int i = 0; i < M_; ++i) { v[i] = expf(v[i] - mx); s += v[i]; }
    float inv = 1.f / s;
    float st = 0.f;
#pragma unroll
    for (int i = 0; i < M_; ++i) {
        float pp = v[i] * inv;                                  // softmax prob
        float g  = 1.f / (1.f + expf(-(pp - 0.02f) * 10.f));    // sigmoid((p-δ)/τ)
        v[i] = pp * g;
        st  += v[i];
    }
    st += 1e-8f;
    float sq = 0.f;
#pragma unroll
    for (int i = 0; i < M_; ++i) { v[i] = v[i] / st + 1e-8f; sq += v[i]; }
    float invq = 1.f / sq;
    float* o = out + (size_t)r * M_;
#pragma unroll
    for (int i = 0; i < M_; ++i) o[i] = v[i] * invq;
}

// ---------------------------------------------------------------------------
extern "C" void kernel_launch(void* const* d_in, const int* in_sizes, int n_in,
                              void* d_out, int out_size, void* d_ws, size_t ws_size,
                              hipStream_t stream) {
    const float* lat = (const float*)d_in[0];   // mixture_latent [B,T,M*5]
    const float* ptw = (const float*)d_in[1];   // per_traj_weight [B,M]
    const float* psw = (const float*)d_in[2];   // per_step_weight [B,T*M]

    float* out        = (float*)d_out;
    float* probs_traj = out + O_TRAJ;
    float* loc        = out + O_LOC;
    float* chol_full  = out + O_CHOL;      // float4-aligned (offset % 4 == 0)
    float* step_probs = out + O_STEP;
    float* mu         = out + O_MU;
    float* chol_step  = out + O_CHOLSTEP;

    // 1) async zero stream of chol_full (must precede the diag scatter)
    //    18,432,000 float4s = 3600 blocks * 5120 float4s/block (exact)
    fill_chol_zero<<<3600, 256, 0, stream>>>((float4*)chol_full);
    // 2) fused transform (writes diag entries on top of zeros)
    transform_kernel<<<N_ELEM / 256, 256, 0, stream>>>(lat, loc, chol_full, mu, chol_step);
    // 3) per-step softmax: B*T rows of width M
    prune_softmax_kernel<<<(B_ * T_ + 255) / 256, 256, 0, stream>>>(psw, step_probs, B_ * T_);
    // 4) per-traj softmax: B rows of width M
    prune_softmax_kernel<<<(B_ + 255) / 256, 256, 0, stream>>>(ptw, probs_traj, B_);
}